// BPFeedForward_7919919693923
// MI455X (gfx1250) — compile-verified
//
#include <hip/hip_runtime.h>

// ---------------------------------------------------------------------------
// Behler-Parrinello per-element MLP, fused, fp32 WMMA (V_WMMA_F32_16X16X4_F32)
//   out[e + 4*m] = Wout[e]^T tanh(W2[e]^T tanh(W1[e]^T tanh(W0[e]^T fps_e[m])))
// E=4, D=128, H=256. One workgroup = 64 atoms of one element, 8 waves.
// Wave tiling: each wave owns ALL 4 m-tiles x 2 n-tiles -> every weight
// fragment is loaded exactly once per workgroup (no intra-WG redundancy),
// 1 memory instruction per WMMA. Activations ping-pong in LDS; weights are
// L2-resident (2.6 MB << 192 MB). fps tile staged with CDNA5 async
// global->LDS copies (GLOBAL_LOAD_ASYNC_TO_LDS_B128, ASYNCcnt-tracked).
// ---------------------------------------------------------------------------

typedef __attribute__((ext_vector_type(2))) float v2f;
typedef __attribute__((ext_vector_type(8))) float v8f;
typedef __attribute__((ext_vector_type(4))) int   v4i;

#define DDIM   128
#define HDIM   256
#define TILE_M 64
#define LDSW   260   // 256 + 4 pad: LDS bank = (4*row + k) % 64, conflict-free
                     // row stride = 1040 B = 65*16 -> keeps 16B alignment

#if __has_builtin(__builtin_amdgcn_global_load_async_to_lds_b128)
#define HAVE_ASYNC_LDS 1
typedef __attribute__((address_space(1))) v4i* gbl_v4i;   // pointer to AS(1) int4
typedef __attribute__((address_space(3))) v4i* lds_v4i;   // pointer to AS(3) int4
#else
#define HAVE_ASYNC_LDS 0
#endif

__device__ __forceinline__ float tanh_act(float x) {
#if __has_builtin(__builtin_amdgcn_tanhf)
  return __builtin_amdgcn_tanhf(x);   // gfx1250 V_TANH_F32 (transcendental unit)
#else
  return tanhf(x);
#endif
}

// One dense layer: sout[64][256] = tanh(sin[64][K] @ W[K][256] + bias).
// WMMA f32 16x16x4 fragment layouts (ISA 7.12.2):
//   A: lane(0-15) row=lane, k+{0,1}; lane(16-31) row=lane-16, k+{2,3}
//   B: lanes(0-15) col=lane, rows ka,ka+1 in v0,v1 (ka = k + 2*(lane>=16))
//   C: vgpr r -> row r (lanes 0-15) / row r+8 (lanes 16-31), col = lane%16
template <int K>
__device__ __forceinline__ void mlp_layer(const float* __restrict__ W,
                                          const float* __restrict__ bias,
                                          const float* __restrict__ sin,
                                          float* __restrict__ sout,
                                          int lane, int wave) {
  const int l16  = lane & 15;
  const int half = lane >> 4;
  const int nt0  = wave * 2;          // 2 n-tiles per wave, 8 waves cover 16

  v8f acc[4][2] = {};                 // 4 m-tiles x 2 n-tiles
  const float* a0 = sin + l16 * LDSW;

  for (int k = 0; k < K; k += 4) {
    const int ka = k + half * 2;
    v2f a[4];
#pragma unroll
    for (int mt = 0; mt < 4; ++mt) {  // ds_load_b64, conflict-free
      const float2 av = *reinterpret_cast<const float2*>(a0 + mt * 16 * LDSW + ka);
      a[mt].x = av.x; a[mt].y = av.y;
    }
    v2f b[2];
#pragma unroll
    for (int ntp = 0; ntp < 2; ++ntp) {  // 64B coalesced per half-wave, L2-hot
      const int col = (nt0 + ntp) * 16 + l16;
      b[ntp].x = W[ka * HDIM + col];
      b[ntp].y = W[(ka + 1) * HDIM + col];
    }
#pragma unroll
    for (int mt = 0; mt < 4; ++mt)
#pragma unroll
      for (int ntp = 0; ntp < 2; ++ntp)
        acc[mt][ntp] = __builtin_amdgcn_wmma_f32_16x16x4_f32(
            /*neg_a=*/false, a[mt], /*neg_b=*/false, b[ntp],
            /*c_mod=*/(short)0, acc[mt][ntp],
            /*reuse_a=*/false, /*reuse_b=*/false);
  }

#pragma unroll
  for (int ntp = 0; ntp < 2; ++ntp) {
    const int col = (nt0 + ntp) * 16 + l16;
    const float bb = bias[col];
#pragma unroll
    for (int mt = 0; mt < 4; ++mt)
#pragma unroll
      for (int r = 0; r < 8; ++r) {
        const int row = mt * 16 + half * 8 + r;
        sout[row * LDSW + col] = tanh_act(acc[mt][ntp][r] + bb);
      }
  }
}

__global__ __launch_bounds__(256) void bp_mlp_kernel(
    const float* __restrict__ fps0, const float* __restrict__ fps1,
    const float* __restrict__ fps2, const float* __restrict__ fps3,
    const float* __restrict__ W0, const float* __restrict__ b0,
    const float* __restrict__ W1, const float* __restrict__ b1,
    const float* __restrict__ W2, const float* __restrict__ b2,
    const float* __restrict__ Wout, float* __restrict__ out, int Mrows) {
  __shared__ alignas(16) float sA[TILE_M * LDSW];   // 66.5 KB
  __shared__ alignas(16) float sB[TILE_M * LDSW];   // 66.5 KB (133 KB < 320 KB)

  const int e    = blockIdx.y;
  const int m0   = blockIdx.x * TILE_M;
  const int tid  = threadIdx.x;
  const int lane = tid & 31;
  const int wave = tid >> 5;

  const float* fps = (e == 0) ? fps0 : (e == 1) ? fps1 : (e == 2) ? fps2 : fps3;
  const float* W0e = W0 + e * DDIM * HDIM;
  const float* W1e = W1 + e * HDIM * HDIM;
  const float* W2e = W2 + e * HDIM * HDIM;
  const float* b0e = b0 + e * HDIM;
  const float* b1e = b1 + e * HDIM;
  const float* b2e = b2 + e * HDIM;
  const float* woe = Wout + e * HDIM;

  // Stage fps tile into sB[.., 0..127]: 16B per lane per step, clamp ragged rows.
  for (int i = tid; i < TILE_M * (DDIM / 4); i += 256) {
    const int r  = i / (DDIM / 4);
    const int c4 = i % (DDIM / 4);
    int gr = m0 + r;
    if (gr >= Mrows) gr = Mrows - 1;
    const float* gsrc = fps + (size_t)gr * DDIM + c4 * 4;
    float*       ldst = sB + r * LDSW + c4 * 4;
#if HAVE_ASYNC_LDS
    // GLOBAL_LOAD_ASYNC_TO_LDS_B128: memory -> LDS, tracked with ASYNCcnt,
    // no VGPR round-trip. Low 32 bits of a generic LDS pointer == AS(3) addr.
    __builtin_amdgcn_global_load_async_to_lds_b128(
        (gbl_v4i)(uintptr_t)gsrc,
        (lds_v4i)(uint32_t)(uintptr_t)ldst,
        /*offset=*/0, /*cpol=*/0);
#else
    const float4 v = *reinterpret_cast<const float4*>(gsrc);
    ldst[0] = v.x; ldst[1] = v.y; ldst[2] = v.z; ldst[3] = v.w;
#endif
  }
#if HAVE_ASYNC_LDS
#if __has_builtin(__builtin_amdgcn_s_wait_asynccnt)
  __builtin_amdgcn_s_wait_asynccnt(0);
#else
  asm volatile("s_wait_asynccnt 0x0" ::: "memory");
#endif
#endif
  __syncthreads();

  mlp_layer<DDIM>(W0e, b0e, sB, sA, lane, wave);
  __syncthreads();
  mlp_layer<HDIM>(W1e, b1e, sA, sB, lane, wave);
  __syncthreads();
  mlp_layer<HDIM>(W2e, b2e, sB, sA, lane, wave);
  __syncthreads();

  // Final layer (H x 1) + scatter back to atom order: atom index = e + 4*m
  if (tid < TILE_M) {
    const float* hr = sA + tid * LDSW;
    float s = 0.f;
#pragma unroll 8
    for (int k = 0; k < HDIM; ++k) s += hr[k] * woe[k];
    const int gm = m0 + tid;
    if (gm < Mrows) out[e + 4 * gm] = s;
  }
}

extern "C" void kernel_launch(void* const* d_in, const int* in_sizes, int n_in,
                              void* d_out, int out_size, void* d_ws, size_t ws_size,
                              hipStream_t stream) {
  const float* fps0 = (const float*)d_in[0];
  const float* fps1 = (const float*)d_in[1];
  const float* fps2 = (const float*)d_in[2];
  const float* fps3 = (const float*)d_in[3];
  const float* W0   = (const float*)d_in[4];
  const float* b0   = (const float*)d_in[5];
  const float* W1   = (const float*)d_in[6];
  const float* b1   = (const float*)d_in[7];
  const float* W2   = (const float*)d_in[8];
  const float* b2   = (const float*)d_in[9];
  const float* Wout = (const float*)d_in[10];
  float* out = (float*)d_out;

  const int Mrows = in_sizes[0] / DDIM;          // 50000
  const int tiles = (Mrows + TILE_M - 1) / TILE_M;
  dim3 grid(tiles, 4);
  bp_mlp_kernel<<<grid, 256, 0, stream>>>(fps0, fps1, fps2, fps3,
                                          W0, b0, W1, b1, W2, b2, Wout,
                                          out, Mrows);
}